// GPStream_69501160784055
// MI455X (gfx1250) — compile-verified
//
#include <hip/hip_runtime.h>
#include <math.h>

// ---------------- problem constants (from reference) ----------------
#define M_TASKS 128
#define NCv     512
#define NTv     256
#define CHUNKv  64
#define NSTEPP  10
#define TOTAL_STEPS 80         // (NC/CHUNK)*N_STEPS
#define LRATE   0.05f
#define BETA1   0.9f
#define BETA2   0.999f
#define EPSA    1e-8f
#define NOISE_FLOORv 1e-4f

#define TPB     256            // 8 wave32
#define NWAVES  8

typedef float v2f __attribute__((ext_vector_type(2)));
typedef float v8f __attribute__((ext_vector_type(8)));

__device__ inline float softplus_f(float x) {
    return (x > 20.f) ? x : log1pf(expf(x));
}
__device__ inline float sigmoid_f(float x) {
    return 1.f / (1.f + expf(-x));
}
__device__ inline float sqd2(const float* a, int i, const float* b, int j) {
    float dx = a[2*i]   - b[2*j];
    float dy = a[2*i+1] - b[2*j+1];
    return dx*dx + dy*dy;
}

// block-wide sum; red is TPB floats of LDS
__device__ float blk_sum(float v, float* red) {
    int tid = threadIdx.x;
    red[tid] = v;
    __syncthreads();
    for (int s = TPB/2; s > 0; s >>= 1) {
        if (tid < s) red[tid] += red[tid + s];
        __syncthreads();
    }
    float r = red[0];
    __syncthreads();
    return r;
}

__device__ inline void tri_decode(int idx, int& bi, int& bj) {
    int b = (int)((sqrtf(8.f * (float)idx + 1.f) - 1.f) * 0.5f);
    while ((b + 1) * (b + 2) / 2 <= idx) b++;
    while (b * (b + 1) / 2 > idx) b--;
    bi = b;
    bj = idx - b * (b + 1) / 2;
}

// Forward solve L z = v   (v in LDS, length n, L row-major stride n, lower)
__device__ void solve_fwd(const float* __restrict__ L, int n, float* v) {
    int tid = threadIdx.x;
    for (int p = 0; p < n; p++) {
        if (tid == 0) v[p] = v[p] / L[(size_t)p * n + p];
        __syncthreads();
        float vp = v[p];
        for (int i = p + 1 + tid; i < n; i += TPB)
            v[i] -= L[(size_t)i * n + p] * vp;
        __syncthreads();
    }
}

// Backward solve L^T a = v  (in place)
__device__ void solve_bwd(const float* __restrict__ L, int n, float* v) {
    int tid = threadIdx.x;
    for (int p = n - 1; p >= 0; p--) {
        if (tid == 0) v[p] = v[p] / L[(size_t)p * n + p];
        __syncthreads();
        float vp = v[p];
        const float* Lp = L + (size_t)p * n;
        for (int i = tid; i < p; i += TPB)
            v[i] -= Lp[i] * vp;
        __syncthreads();
    }
}

// In-place blocked Cholesky (lower), panel width 16, WMMA f32 trailing SYRK.
// A: row-major n x n (n multiple of 16). panel: (NC-16)*17 LDS, dblk: 16*17 LDS.
__device__ void chol_inplace(float* __restrict__ A, int n,
                             float* panel, float* dblk) {
    const int tid  = threadIdx.x;
    const int lane = tid & 31;
    const int wv   = tid >> 5;
    const int l    = lane & 15;
    const int hi   = lane >> 4;   // 0 or 1

    const int nb = n / 16;
    for (int kb = 0; kb < nb; kb++) {
        const int k0 = kb * 16;
        // --- load 16x16 diagonal block into LDS ---
        {
            int i = tid >> 4, j = tid & 15;
            dblk[i * 17 + j] = A[(size_t)(k0 + i) * n + (k0 + j)];
        }
        __syncthreads();
        // --- factor diagonal block (tiny: serial on thread 0) ---
        if (tid == 0) {
            for (int j = 0; j < 16; j++) {
                float djj = sqrtf(dblk[j * 17 + j]);
                dblk[j * 17 + j] = djj;
                float inv = 1.f / djj;
                for (int i = j + 1; i < 16; i++) dblk[i * 17 + j] *= inv;
                for (int i = j + 1; i < 16; i++) {
                    float lij = dblk[i * 17 + j];
                    for (int p = j + 1; p <= i; p++)
                        dblk[i * 17 + p] -= lij * dblk[p * 17 + j];
                }
            }
        }
        __syncthreads();
        // --- write factored diag block back ---
        {
            int i = tid >> 4, j = tid & 15;
            A[(size_t)(k0 + i) * n + (k0 + j)] = dblk[i * 17 + j];
        }
        // --- panel triangular solve: L[i, k0:k0+16] = A * inv(Lkk^T) ---
        const int m = n - k0 - 16;
        for (int rr = tid; rr < m; rr += TPB) {
            int gi = k0 + 16 + rr;
            float x[16];
            for (int j = 0; j < 16; j++) x[j] = A[(size_t)gi * n + k0 + j];
            for (int j = 0; j < 16; j++) {
                float acc = x[j];
                for (int p = 0; p < j; p++) acc -= x[p] * dblk[j * 17 + p];
                x[j] = acc / dblk[j * 17 + j];
            }
            for (int j = 0; j < 16; j++) {
                A[(size_t)gi * n + k0 + j] = x[j];
                panel[rr * 17 + j] = x[j];
            }
        }
        __syncthreads();
        // --- trailing update (lower tiles): C -= Pi * Pj^T via WMMA f32 ---
        const int mb = m / 16;
        const int ntile = mb * (mb + 1) / 2;
        for (int idx = wv; idx < ntile; idx += NWAVES) {
            int bi, bj;
            tri_decode(idx, bi, bj);
            int gr = k0 + 16 + bi * 16;
            int gc = k0 + 16 + bj * 16;
            v8f c;
#pragma unroll
            for (int vv = 0; vv < 8; vv++)
                c[vv] = A[(size_t)(gr + vv + 8 * hi) * n + gc + l];
#pragma unroll
            for (int q = 0; q < 4; q++) {
                int ko = 4 * q + 2 * hi;
                v2f a, b;
                a.x = -panel[(bi * 16 + l) * 17 + ko];
                a.y = -panel[(bi * 16 + l) * 17 + ko + 1];
                b.x =  panel[(bj * 16 + l) * 17 + ko];
                b.y =  panel[(bj * 16 + l) * 17 + ko + 1];
                c = __builtin_amdgcn_wmma_f32_16x16x4_f32(
                        false, a, false, b, (short)0, c, false, false);
            }
#pragma unroll
            for (int vv = 0; vv < 8; vv++)
                A[(size_t)(gr + vv + 8 * hi) * n + gc + l] = c[vv];
        }
        __syncthreads();
    }
}

__global__ __launch_bounds__(TPB)
void gp_full_kernel(const float* __restrict__ xc,
                    const float* __restrict__ yc,
                    const float* __restrict__ xtg,
                    float* __restrict__ out,
                    float* __restrict__ ws,
                    int nslots) {
    __shared__ float xs[NCv * 2];
    __shared__ float xts[NTv * 2];
    __shared__ float ysh[NCv];
    __shared__ float av[NCv];
    __shared__ float zv[NCv];
    __shared__ float red[TPB];
    __shared__ float dblk[16 * 17];
    __shared__ float panel[(NCv - 16) * 17];
    __shared__ float par[4], mAd[4], vAd[4];

    const int tid  = threadIdx.x;
    const int lane = tid & 31;
    const int wv   = tid >> 5;
    const int l    = lane & 15;
    const int hi   = lane >> 4;

    float* Kb = ws + (size_t)blockIdx.x * (2ull * NCv * NCv);
    float* Wb = Kb + (size_t)NCv * NCv;

    for (int task = blockIdx.x; task < M_TASKS; task += nslots) {
        // ------- load task data to LDS -------
        for (int i = tid; i < NCv * 2; i += TPB) xs[i]  = xc[(size_t)task * NCv * 2 + i];
        for (int i = tid; i < NCv;     i += TPB) ysh[i] = yc[(size_t)task * NCv + i];
        for (int i = tid; i < NTv * 2; i += TPB) xts[i] = xtg[(size_t)task * NTv * 2 + i];
        if (tid < 4) { par[tid] = 0.f; mAd[tid] = 0.f; vAd[tid] = 0.f; }
        __syncthreads();

        // ================= Adam hyperparameter loop =================
        for (int t = 0; t < TOTAL_STEPS; t++) {
            const int n = ((t / NSTEPP) + 1) * CHUNKv;   // 64..512, mult of 16
            const float nf = (float)n;
            const float mean_c = par[0];
            const float ell    = softplus_f(par[1]);
            const float s      = softplus_f(par[2]);
            const float noise  = softplus_f(par[3]) + NOISE_FLOORv;
            const float h      = 0.5f / (ell * ell);

            // build active-block kernel matrix K (stride n)
            for (int idx = tid; idx < n * n; idx += TPB) {
                int i = idx / n, j = idx - i * n;
                float d = sqd2(xs, i, xs, j);
                Kb[idx] = s * expf(-d * h) + ((i == j) ? noise : 0.f);
            }
            __syncthreads();

            chol_inplace(Kb, n, panel, dblk);

            // alpha = K^{-1} r
            for (int i = tid; i < n; i += TPB) zv[i] = ysh[i] - mean_c;
            __syncthreads();
            solve_fwd(Kb, n, zv);
            solve_bwd(Kb, n, zv);
            for (int i = tid; i < n; i += TPB) av[i] = zv[i];
            __syncthreads();

            // scalar sums
            float psa = 0.f, pq = 0.f, pn2 = 0.f;
            for (int i = tid; i < n; i += TPB) {
                float a = av[i];
                psa += a; pn2 += a * a; pq += (ysh[i] - mean_c) * a;
            }
            float sa   = blk_sum(psa, red);
            float quad = blk_sum(pq,  red);
            float na2  = blk_sum(pn2, red);

            // Linv = L^{-1} (lower), into Wb (row-major stride n)
            for (int idx = tid; idx < n * n; idx += TPB) Wb[idx] = 0.f;
            __syncthreads();
            for (int j = tid; j < n; j += TPB) {
                Wb[(size_t)j * n + j] = 1.f / Kb[(size_t)j * n + j];
                for (int k = j + 1; k < n; k++) {
                    const float* Lk = Kb + (size_t)k * n;
                    float acc = 0.f;
                    for (int p = j; p < k; p++) acc += Lk[p] * Wb[(size_t)p * n + j];
                    Wb[(size_t)k * n + j] = -acc / Lk[k];
                }
            }
            __syncthreads();

            // Kinv (lower tiles) = Linv^T * Linv via WMMA SYRK -> overwrite Kb
            {
                int nb = n / 16;
                int ntile = nb * (nb + 1) / 2;
                for (int idx = wv; idx < ntile; idx += NWAVES) {
                    int bi, bj;
                    tri_decode(idx, bi, bj);
                    v8f c;
#pragma unroll
                    for (int vv = 0; vv < 8; vv++) c[vv] = 0.f;
                    for (int k4 = bi * 16; k4 < n; k4 += 4) {   // k >= i rows only
                        v2f a, b;
                        a.x = Wb[(size_t)(k4 + 2 * hi)     * n + bi * 16 + l];
                        a.y = Wb[(size_t)(k4 + 2 * hi + 1) * n + bi * 16 + l];
                        b.x = Wb[(size_t)(k4 + 2 * hi)     * n + bj * 16 + l];
                        b.y = Wb[(size_t)(k4 + 2 * hi + 1) * n + bj * 16 + l];
                        c = __builtin_amdgcn_wmma_f32_16x16x4_f32(
                                false, a, false, b, (short)0, c, false, false);
                    }
#pragma unroll
                    for (int vv = 0; vv < 8; vv++)
                        Kb[(size_t)(bi * 16 + vv + 8 * hi) * n + bj * 16 + l] = c[vv];
                }
            }
            __syncthreads();

            // tr(Kinv)
            float ptr = 0.f;
            for (int i = tid; i < n; i += TPB) ptr += Kb[(size_t)i * n + i];
            float trK = blk_sum(ptr, red);

            // A1 = sum Kinv.*Kse.*D ; A2 = sum (aa^T).*Kse.*D  (diag D=0)
            float a1p = 0.f, a2p = 0.f;
            for (int i = 1; i < n; i++) {
                float xi0 = xs[2 * i], xi1 = xs[2 * i + 1], ai = av[i];
                for (int j = tid; j < i; j += TPB) {
                    float dx = xi0 - xs[2 * j], dy = xi1 - xs[2 * j + 1];
                    float d = dx * dx + dy * dy;
                    float w = 2.f * s * expf(-d * h) * d;
                    a1p += Kb[(size_t)i * n + j] * w;
                    a2p += ai * av[j] * w;
                }
            }
            float A1 = blk_sum(a1p, red);
            float A2 = blk_sum(a2p, red);

            // closed-form gradients + softplus chain + Adam
            float gm  = -sa / nf;
            float gsp = 0.5f / (nf * s) * (nf - noise * trK - quad + noise * na2);
            float glp = 0.5f / nf * (A1 - A2) / (ell * ell * ell);
            float gnp = 0.5f / nf * (trK - na2);
            if (tid == 0) {
                float g[4];
                g[0] = gm;
                g[1] = glp * sigmoid_f(par[1]);
                g[2] = gsp * sigmoid_f(par[2]);
                g[3] = gnp * sigmoid_f(par[3]);
                float tf  = (float)(t + 1);
                float bc1 = 1.f - powf(BETA1, tf);
                float bc2 = 1.f - powf(BETA2, tf);
                for (int k = 0; k < 4; k++) {
                    mAd[k] = BETA1 * mAd[k] + (1.f - BETA1) * g[k];
                    vAd[k] = BETA2 * vAd[k] + (1.f - BETA2) * g[k] * g[k];
                    par[k] -= LRATE * (mAd[k] / bc1) / (sqrtf(vAd[k] / bc2) + EPSA);
                }
            }
            __syncthreads();
        }

        // ================= final prediction (n = 512) =================
        {
            const int n = NCv;
            const float mean_c = par[0];
            const float ell    = softplus_f(par[1]);
            const float s      = softplus_f(par[2]);
            const float noise  = softplus_f(par[3]) + NOISE_FLOORv;
            const float h      = 0.5f / (ell * ell);

            for (int idx = tid; idx < n * n; idx += TPB) {
                int i = idx >> 9, j = idx & (NCv - 1);
                float d = sqd2(xs, i, xs, j);
                Kb[idx] = s * expf(-d * h) + ((i == j) ? noise : 0.f);
            }
            __syncthreads();
            chol_inplace(Kb, n, panel, dblk);

            for (int i = tid; i < n; i += TPB) zv[i] = ysh[i] - mean_c;
            __syncthreads();
            solve_fwd(Kb, n, zv);
            solve_bwd(Kb, n, zv);
            for (int i = tid; i < n; i += TPB) av[i] = zv[i];
            __syncthreads();

            // Ktc[q][i] into Wb (stride NC)
            for (int idx = tid; idx < NTv * NCv; idx += TPB) {
                int q = idx >> 9, i = idx & (NCv - 1);
                float d = sqd2(xts, q, xs, i);
                Wb[idx] = s * expf(-d * h);
            }
            __syncthreads();

            // mean = mean_c + Ktc @ alpha  (one target per thread)
            {
                int q = tid;   // TPB == NT
                const float* kt = Wb + (size_t)q * NCv;
                float acc = 0.f;
                for (int i = 0; i < NCv; i++) acc += kt[i] * av[i];
                out[(size_t)task * NTv + q] = mean_c + acc;
            }

            // U = L^{-1} Ktc^T, stored [i][q] (stride NT), one column per thread
            float* Ub = Wb + (size_t)NTv * NCv;
            {
                int q = tid;
                for (int i = 0; i < NCv; i++) {
                    const float* Li = Kb + (size_t)i * NCv;
                    float acc = Wb[(size_t)q * NCv + i];
                    for (int p = 0; p < i; p++) acc -= Li[p] * Ub[(size_t)p * NTv + q];
                    Ub[(size_t)i * NTv + q] = acc / Li[i];
                }
            }
            __syncthreads();

            // cov = Ktt + noise*I - U^T U  via WMMA SYRK over K=512
            size_t covbase = (size_t)M_TASKS * NTv + (size_t)task * NTv * NTv;
            {
                int nb = NTv / 16;
                int ntile = nb * (nb + 1) / 2;
                for (int idx = wv; idx < ntile; idx += NWAVES) {
                    int bi, bj;
                    tri_decode(idx, bi, bj);
                    v8f c;
#pragma unroll
                    for (int vv = 0; vv < 8; vv++) c[vv] = 0.f;
                    for (int k4 = 0; k4 < NCv; k4 += 4) {
                        v2f a, b;
                        a.x = Ub[(size_t)(k4 + 2 * hi)     * NTv + bi * 16 + l];
                        a.y = Ub[(size_t)(k4 + 2 * hi + 1) * NTv + bi * 16 + l];
                        b.x = Ub[(size_t)(k4 + 2 * hi)     * NTv + bj * 16 + l];
                        b.y = Ub[(size_t)(k4 + 2 * hi + 1) * NTv + bj * 16 + l];
                        c = __builtin_amdgcn_wmma_f32_16x16x4_f32(
                                false, a, false, b, (short)0, c, false, false);
                    }
#pragma unroll
                    for (int vv = 0; vv < 8; vv++) {
                        int q1 = bi * 16 + vv + 8 * hi;
                        int q2 = bj * 16 + l;
                        float d = sqd2(xts, q1, xts, q2);
                        float val = s * expf(-d * h)
                                  + ((q1 == q2) ? noise : 0.f) - c[vv];
                        out[covbase + (size_t)q1 * NTv + q2] = val;
                        if (bi != bj)
                            out[covbase + (size_t)q2 * NTv + q1] = val;
                    }
                }
            }
            __syncthreads();
        }
        __syncthreads();   // LDS reuse fence before next task
    }
}

extern "C" void kernel_launch(void* const* d_in, const int* in_sizes, int n_in,
                              void* d_out, int out_size, void* d_ws, size_t ws_size,
                              hipStream_t stream) {
    const float* xc = (const float*)d_in[0];   // (128,512,2)
    const float* yc = (const float*)d_in[1];   // (128,512,1)
    const float* xt = (const float*)d_in[2];   // (128,256,2)
    float* out = (float*)d_out;                // means (128*256) ++ covs (128*256*256)
    float* ws  = (float*)d_ws;

    const size_t slot_bytes = (size_t)2 * NCv * NCv * sizeof(float);  // 2 MB/slot
    int nslots = (int)(ws_size / slot_bytes);
    if (nslots < 1) nslots = 1;
    if (nslots > M_TASKS) nslots = M_TASKS;

    gp_full_kernel<<<dim3(nslots), dim3(TPB), 0, stream>>>(xc, yc, xt, out, ws, nslots);
}